// SemiGraphConv_59390807769609
// MI455X (gfx1250) — compile-verified
//
#include <hip/hip_runtime.h>
#include <hip/hip_bf16.h>

typedef __attribute__((ext_vector_type(2))) float v2f;
typedef __attribute__((ext_vector_type(8))) float v8f;

#define D_IN  256
#define D_OUT 128
#define LDSS  260   // padded LDS row stride (dwords): 260 % 64 = 4 -> conflict-free A-frag b64 reads
#define ROWS  32    // row tile per block (two 16-row WMMA tiles per wave -> B-fragment reuse x2)

// ---------------------------------------------------------------------------
// xl = x @ W + b  via V_WMMA_F32_16X16X4_F32.
// Block = 256 threads = 8 waves; block computes rows [row0, row0+32) x all 128 cols.
// Wave w owns columns [16w, 16w+16) with TWO accumulators (rows 0-15 / 16-31 of
// the tile) sharing each B fragment: 2 independent WMMAs per B load.
// A-tile (32x256) staged once in LDS; B fragments stream from global
// (W = 128KB, permanently L0/L2-resident).
// Epilogue also writes out = mask ? 0 : xl (so the scatter phase is add-only).
// ---------------------------------------------------------------------------
__global__ __launch_bounds__(256) void gcn_gemm_wmma(
    const float* __restrict__ x, const float* __restrict__ W,
    const float* __restrict__ bias, const unsigned char* __restrict__ mask,
    float* __restrict__ xl, float* __restrict__ out, int n_nodes)
{
    __shared__ float lx[ROWS * LDSS];
    const int tid  = threadIdx.x;
    const int row0 = blockIdx.x * ROWS;

    // Stage 32x256 fp32 tile of x into LDS, coalesced float4 loads.
    for (int i = tid; i < ROWS * (D_IN / 4); i += 256) {
        const int r  = i >> 6;            // 64 float4 per row
        const int c4 = (i & 63) << 2;
        float4 v = {0.f, 0.f, 0.f, 0.f};
        const int row = row0 + r;
        if (row < n_nodes) v = *(const float4*)(x + (size_t)row * D_IN + c4);
        *(float4*)(&lx[r * LDSS + c4]) = v;   // 1040B row stride keeps 16B alignment
    }
    __syncthreads();

    const int wave = tid >> 5;
    const int lane = tid & 31;
    const int l16  = lane & 15;
    const int hi   = lane >> 4;           // lo half: K+{0,1}; hi half: K+{2,3}
    const int n0   = wave * 16;

    v8f acc0 = {};                        // tile rows 0..15
    v8f acc1 = {};                        // tile rows 16..31
    // A fragment sources: row l16 (+16) of LDS tile, K offset 2*hi
    const float* __restrict__ ap0 = &lx[l16 * LDSS + 2 * hi];
    const float* __restrict__ ap1 = ap0 + 16 * LDSS;
    // B fragment source: W[k + 2*hi][n0 + l16]
    const float* __restrict__ wp  = W + n0 + l16 + (size_t)(2 * hi) * D_OUT;

    #pragma unroll 4
    for (int k0 = 0; k0 < D_IN; k0 += 4) {
        v2f b;  b.x  = wp [(size_t)k0 * D_OUT];
                b.y  = wp [(size_t)(k0 + 1) * D_OUT];
        v2f a0; a0.x = ap0[k0];
                a0.y = ap0[k0 + 1];
        v2f a1; a1.x = ap1[k0];
                a1.y = ap1[k0 + 1];
        // two independent v_wmma_f32_16x16x4_f32 sharing the B fragment
        acc0 = __builtin_amdgcn_wmma_f32_16x16x4_f32(
                   false, a0, false, b, (short)0, acc0, false, false);
        acc1 = __builtin_amdgcn_wmma_f32_16x16x4_f32(
                   false, a1, false, b, (short)0, acc1, false, false);
    }

    // C/D layout: VGPR j -> M = j + 8*hi (+0 / +16 for acc0/acc1), N = l16 (+ n0)
    const int   ncol = n0 + l16;
    const float bv   = bias[ncol];
    const bool  mk   = mask[ncol] != 0;
    float* __restrict__ xlp  = xl  + (size_t)(row0 + 8 * hi) * D_OUT + ncol;
    float* __restrict__ outp = out + (size_t)(row0 + 8 * hi) * D_OUT + ncol;

    if (row0 + ROWS <= n_nodes) {
        // fast path: full tile in range, straight-line stores
        #pragma unroll
        for (int j = 0; j < 8; ++j) {
            const float v0 = acc0[j] + bv;
            const float v1 = acc1[j] + bv;
            xlp [(size_t)j        * D_OUT] = v0;
            xlp [(size_t)(j + 16) * D_OUT] = v1;
            outp[(size_t)j        * D_OUT] = mk ? 0.f : v0;
            outp[(size_t)(j + 16) * D_OUT] = mk ? 0.f : v1;
        }
    } else {
        // ragged last block
        #pragma unroll
        for (int j = 0; j < 8; ++j) {
            const int r0 = row0 + 8 * hi + j;
            const int r1 = r0 + 16;
            if (r0 < n_nodes) {
                const float v0 = acc0[j] + bv;
                xlp [(size_t)j * D_OUT] = v0;
                outp[(size_t)j * D_OUT] = mk ? 0.f : v0;
            }
            if (r1 < n_nodes) {
                const float v1 = acc1[j] + bv;
                xlp [(size_t)(j + 16) * D_OUT] = v1;
                outp[(size_t)(j + 16) * D_OUT] = mk ? 0.f : v1;
            }
        }
    }
}

// ---------------------------------------------------------------------------
// deg[src[e]] += 1
// ---------------------------------------------------------------------------
__global__ __launch_bounds__(256) void gcn_degree(
    const int* __restrict__ src, float* __restrict__ deg, int n_edges)
{
    const int e = blockIdx.x * 256 + threadIdx.x;
    if (e < n_edges) atomicAdd(&deg[src[e]], 1.0f);
}

// ---------------------------------------------------------------------------
// One wave32 per edge; lane owns 4 channels (float4 gather, masked atomicAdd).
// xl and out are both L2-resident (25.6MB each), so this runs at L2 speed.
// ---------------------------------------------------------------------------
__global__ __launch_bounds__(256) void gcn_scatter(
    const int* __restrict__ src, const int* __restrict__ dst,
    const float* __restrict__ deg, const float* __restrict__ xl,
    const unsigned char* __restrict__ mask, float* __restrict__ out, int n_edges)
{
    const int lane = threadIdx.x & 31;
    const int e    = (blockIdx.x * 256 + threadIdx.x) >> 5;
    const int c    = lane << 2;
    // Per-lane channel mask is loop-invariant (constant per lane).
    const bool m0 = mask[c + 0] != 0;
    const bool m1 = mask[c + 1] != 0;
    const bool m2 = mask[c + 2] != 0;
    const bool m3 = mask[c + 3] != 0;

    if (e >= n_edges) return;
    const int s = src[e];
    const int d = dst[e];
    const float coef = rsqrtf(fmaxf(deg[s], 1.f) * fmaxf(deg[d], 1.f));
    const float4 v = *(const float4*)(xl + (size_t)s * D_OUT + c);
    float* op = out + (size_t)d * D_OUT + c;
    if (m0) atomicAdd(op + 0, v.x * coef);
    if (m1) atomicAdd(op + 1, v.y * coef);
    if (m2) atomicAdd(op + 2, v.z * coef);
    if (m3) atomicAdd(op + 3, v.w * coef);
}

// ---------------------------------------------------------------------------
extern "C" void kernel_launch(void* const* d_in, const int* in_sizes, int n_in,
                              void* d_out, int out_size, void* d_ws, size_t ws_size,
                              hipStream_t stream)
{
    const float*         x    = (const float*)d_in[0];
    const float*         W    = (const float*)d_in[1];
    const float*         b    = (const float*)d_in[2];
    const int*           src  = (const int*)d_in[3];
    const int*           dst  = (const int*)d_in[4];
    const unsigned char* mask = (const unsigned char*)d_in[5];  // numpy bool = 1 byte

    const int n_nodes = in_sizes[0] / D_IN;
    const int n_edges = in_sizes[3];

    float* deg = (float*)d_ws;
    float* xl  = deg + (((size_t)n_nodes + 255) & ~(size_t)255);  // 1KB-aligned
    float* out = (float*)d_out;

    hipMemsetAsync(deg, 0, sizeof(float) * (size_t)n_nodes, stream);

    gcn_degree<<<(n_edges + 255) / 256, 256, 0, stream>>>(src, deg, n_edges);

    gcn_gemm_wmma<<<(n_nodes + ROWS - 1) / ROWS, 256, 0, stream>>>(
        x, W, b, mask, xl, out, n_nodes);

    // one wave per edge -> 8 edges per 256-thread block
    gcn_scatter<<<((size_t)n_edges + 7) / 8, 256, 0, stream>>>(
        src, dst, deg, xl, mask, out, n_edges);
}